// CustomSelfAttention_62079457296431
// MI455X (gfx1250) — compile-verified
//
#include <hip/hip_runtime.h>
#include <math.h>

// CDNA5 / gfx1250 fp32 WMMA types
typedef __attribute__((ext_vector_type(2))) float v2f;
typedef __attribute__((ext_vector_type(8))) float v8f;

// Types for the async global->LDS builtins (signatures revealed by clang):
//   b128: (int4 AS1*, lds, imm, imm) ; b64: (int2 AS1*, ...) ; b32: (int AS1*, ...)
typedef __attribute__((__vector_size__(4 * sizeof(int)))) int i32x4;
typedef __attribute__((__vector_size__(2 * sizeof(int)))) int i32x2;
typedef __attribute__((address_space(1))) i32x4* gp128;
typedef __attribute__((address_space(1))) i32x2* gp64;
typedef __attribute__((address_space(1))) int*   gp32;
typedef __attribute__((address_space(3))) i32x4* lp128;
typedef __attribute__((address_space(3))) i32x2* lp64;
typedef __attribute__((address_space(3))) int*   lp32;

#if __has_builtin(__builtin_amdgcn_global_load_async_to_lds_b128) && \
    __has_builtin(__builtin_amdgcn_global_load_async_to_lds_b64)  && \
    __has_builtin(__builtin_amdgcn_global_load_async_to_lds_b32)
#define HAVE_ASYNC_LDS 1
#else
#define HAVE_ASYNC_LDS 0
#endif

// Problem constants (match reference)
#define BATCH 8
#define SEQ   2048
#define EMB   512

#define AS_STRIDE 20   // A tile rows padded: 16B-aligned, bank-conflict-free
#define BS_N      144  // B tile n-dim padded: kpair rows on disjoint bank halves

// ldb is EMB at every call site -> compile-time immediate offsets for the
// strided column loads in the !BT staging path.
#define LDB EMB

// ---------------------------------------------------------------------------
// 128x128-tile fp32 WMMA GEMM:  C = op(A[M,K] x B) (+bias) (*scale)
//   BT      : B accessed transposed: B_logical[k][n] = Bmem[n*LDB + k]
//   USE_BIAS: add bias[n] in epilogue
//   SCALE   : multiply by scale in epilogue
// Block = 256 threads (8 wave32) arranged 4x2; each wave owns a 32x64 patch
// = 2x4 WMMA tiles (8 x v8f accumulators). K staged through ping-pong LDS in
// chunks of 16 (4 WMMA k-steps).
//
// Staging uses GLOBAL_LOAD_ASYNC_TO_LDS (ASYNCcnt): zero staging VGPRs, data
// lands in LDS while WMMAs run on the other buffer -> no spills under the
// 128-VGPR cap of 8-wave workgroups. Fallback: register staging.
//
// B is stored k-pair interleaved in LDS: element (k,n) -> Bs[k/2][n][k&1],
// so each B fragment {B[kk+2h][n], B[kk+2h+1][n]} is ONE aligned ds_load_b64
// straight into an even VGPR pair (no packing movs). In the BT async path
// the interleave maps exactly onto async b64 copies.
// ---------------------------------------------------------------------------
template <bool BT, bool USE_BIAS, bool SCALE>
__global__ __launch_bounds__(256)
void wmma_gemm_128x128(const float* __restrict__ A,
                       const float* __restrict__ Bm,
                       const float* __restrict__ bias,
                       float* __restrict__ C,
                       int K, int lda, int ldc,
                       long aBatch, long bBatch, long cBatch,
                       float scale) {
    const int mBase = blockIdx.x * 128;
    const int nBase = blockIdx.y * 128;
    A  += (long)blockIdx.z * aBatch;
    Bm += (long)blockIdx.z * bBatch;
    C  += (long)blockIdx.z * cBatch;

    __shared__ float As[2][128][AS_STRIDE];
    __shared__ float Bs[2][8][BS_N][2];   // [buf][kpair][n][k&1]

    const int t    = threadIdx.x;      // 0..255
    const int wave = t >> 5;           // 0..7
    const int lane = t & 31;
    const int half = lane >> 4;        // 0: lanes 0-15, 1: lanes 16-31
    const int lr   = lane & 15;
    const int wr   = (wave >> 1) * 32; // wave row offset (0,32,64,96)
    const int wc   = (wave & 1)  * 64; // wave col offset (0,64)

    v8f acc[2][4];
#pragma unroll
    for (int i = 0; i < 2; ++i)
#pragma unroll
        for (int j = 0; j < 4; ++j)
            acc[i][j] = (v8f){0.f,0.f,0.f,0.f,0.f,0.f,0.f,0.f};

    // A tile: 128 rows x 16 cols, half-row per thread (2 x 16B).
    const int aRowIdx = t >> 1;          // 0..127
    const int aCol    = (t & 1) * 8;     // 0 or 8
    const float* aRow = A + (long)(mBase + aRowIdx) * lda + aCol;
    // B tile: thread owns half an n-column: n = t & 127, k in [kh, kh+8).
    const int bN  = t & 127;
    const int bKh = (t >> 7) * 8;        // 0 or 8
    const float* bColPtrT = Bm + (long)(nBase + bN) * LDB + bKh; // BT
    const float* bColPtrN = Bm + (long)bKh * LDB + nBase + bN;   // !BT

#if HAVE_ASYNC_LDS
    // ---- async staging: global -> LDS with no VGPR data movement ----
    auto loadChunkAsync = [&](int k0, int buf) {
#pragma unroll
        for (int q = 0; q < 2; ++q)
            __builtin_amdgcn_global_load_async_to_lds_b128(
                (gp128)(aRow + k0 + 4 * q),
                (lp128)&As[buf][aRowIdx][aCol + 4 * q], 0, 0);
        if (BT) {
            // global pair {B[n][k],B[n][k+1]} -> interleaved LDS slot: b64
#pragma unroll
            for (int q = 0; q < 4; ++q)
                __builtin_amdgcn_global_load_async_to_lds_b64(
                    (gp64)(bColPtrT + k0 + 2 * q),
                    (lp64)&Bs[buf][bKh / 2 + q][bN][0], 0, 0);
        } else {
#pragma unroll
            for (int k = 0; k < 8; ++k)
                __builtin_amdgcn_global_load_async_to_lds_b32(
                    (gp32)(bColPtrN + (long)(k0 + k) * LDB),
                    (lp32)&Bs[buf][(bKh + k) / 2][bN][(bKh + k) & 1], 0, 0);
        }
    };
#else
    // ---- fallback staging: global -> registers -> LDS ----
    float4 ra[2];
    float  rbk[8];

    auto loadChunk = [&](int k0) {
        if (BT) {
#pragma unroll
            for (int q = 0; q < 2; ++q) {
                float4 v = *(const float4*)(bColPtrT + k0 + 4 * q);
                rbk[4 * q + 0] = v.x; rbk[4 * q + 1] = v.y;
                rbk[4 * q + 2] = v.z; rbk[4 * q + 3] = v.w;
            }
        } else {
#pragma unroll
            for (int k = 0; k < 8; ++k)
                rbk[k] = bColPtrN[(long)(k0 + k) * LDB];
        }
#pragma unroll
        for (int q = 0; q < 2; ++q)
            ra[q] = *(const float4*)(aRow + k0 + 4 * q);
    };

    auto storeChunk = [&](int buf) {
#pragma unroll
        for (int q = 0; q < 4; ++q)
            *(v2f*)&Bs[buf][bKh / 2 + q][bN][0] = (v2f){rbk[2 * q], rbk[2 * q + 1]};
#pragma unroll
        for (int q = 0; q < 2; ++q)
            *(float4*)&As[buf][aRowIdx][aCol + 4 * q] = ra[q];
    };
#endif

    auto computeChunk = [&](int buf) {
#pragma unroll
        for (int kk = 0; kk < 16; kk += 4) {
            v2f af[2], bf[4];
            // A frag (16x4): VGPR0/1 = K{kk,kk+1} for lanes 0-15,
            //                K{kk+2,kk+3} for lanes 16-31.
#pragma unroll
            for (int i = 0; i < 2; ++i) {
                const float* p = &As[buf][wr + i * 16 + lr][kk + 2 * half];
                af[i] = (v2f){p[0], p[1]};
            }
            // B frag (4x16): single b64 from interleaved layout, N = lr.
#pragma unroll
            for (int j = 0; j < 4; ++j)
                bf[j] = *(const v2f*)&Bs[buf][kk / 2 + half][wc + j * 16 + lr][0];
#pragma unroll
            for (int i = 0; i < 2; ++i)
#pragma unroll
                for (int j = 0; j < 4; ++j)
                    acc[i][j] = __builtin_amdgcn_wmma_f32_16x16x4_f32(
                        false, af[i], false, bf[j],
                        (short)0, acc[i][j], false, false);
        }
    };

#if HAVE_ASYNC_LDS
    // --- pipeline: async into buf^1 while buf computes; one barrier/iter ---
    loadChunkAsync(0, 0);
    asm volatile("s_wait_asynccnt 0x0" ::: "memory");
    __syncthreads();

    int buf = 0;
    for (int k0 = 16; k0 < K; k0 += 16) {
        loadChunkAsync(k0, buf ^ 1);  // global -> LDS, no VGPRs
        computeChunk(buf);            // 32 WMMAs/wave
        asm volatile("s_wait_asynccnt 0x0" ::: "memory");
        __syncthreads();
        buf ^= 1;
    }
    computeChunk(buf);
#else
    // --- pipeline: prefetch regs -> compute -> store other buffer ---
    loadChunk(0);
    storeChunk(0);
    __syncthreads();

    int buf = 0;
    for (int k0 = 16; k0 < K; k0 += 16) {
        loadChunk(k0);
        computeChunk(buf);
        storeChunk(buf ^ 1);
        __syncthreads();
        buf ^= 1;
    }
    computeChunk(buf);
#endif

    // ---- epilogue: C/D layout (N = lr; M = r + 8*half per VGPR r) ----
#pragma unroll
    for (int i = 0; i < 2; ++i) {
#pragma unroll
        for (int j = 0; j < 4; ++j) {
            const int n = nBase + wc + j * 16 + lr;
            const float bv = USE_BIAS ? bias[n] : 0.0f;
#pragma unroll
            for (int r = 0; r < 8; ++r) {
                const int m = mBase + wr + i * 16 + r + 8 * half;
                float val = acc[i][j][r];
                if (SCALE) val *= scale;
                C[(long)m * ldc + n] = val + bv;
            }
        }
    }
}

// ---------------------------------------------------------------------------
// Row softmax, in place. One 256-thread block per row of length SEQ (=2048).
// ---------------------------------------------------------------------------
__global__ __launch_bounds__(256)
void softmax_rows(float* __restrict__ w) {
    const long row = blockIdx.x;
    float* p = w + row * (long)SEQ;
    const int t = threadIdx.x;

    float vals[SEQ / 256];
    float m = -INFINITY;
#pragma unroll
    for (int i = 0; i < SEQ / 256; ++i) {
        vals[i] = p[t + i * 256];
        m = fmaxf(m, vals[i]);
    }

    __shared__ float red[256];
    red[t] = m;
    __syncthreads();
#pragma unroll
    for (int s = 128; s > 0; s >>= 1) {
        if (t < s) red[t] = fmaxf(red[t], red[t + s]);
        __syncthreads();
    }
    m = red[0];
    __syncthreads();

    float sum = 0.0f;
#pragma unroll
    for (int i = 0; i < SEQ / 256; ++i) {
        vals[i] = __expf(vals[i] - m);
        sum += vals[i];
    }
    red[t] = sum;
    __syncthreads();
#pragma unroll
    for (int s = 128; s > 0; s >>= 1) {
        if (t < s) red[t] += red[t + s];
        __syncthreads();
    }
    const float inv = 1.0f / red[0];
#pragma unroll
    for (int i = 0; i < SEQ / 256; ++i)
        p[t + i * 256] = vals[i] * inv;
}

// ---------------------------------------------------------------------------
extern "C" void kernel_launch(void* const* d_in, const int* in_sizes, int n_in,
                              void* d_out, int out_size, void* d_ws, size_t ws_size,
                              hipStream_t stream) {
    (void)in_sizes; (void)n_in; (void)out_size; (void)ws_size;

    const float* X  = (const float*)d_in[0]; // [B,S,E] -> [B*S, E]
    const float* Wq = (const float*)d_in[1];
    const float* bq = (const float*)d_in[2];
    const float* Wk = (const float*)d_in[3];
    const float* bk = (const float*)d_in[4];
    const float* Wv = (const float*)d_in[5];
    const float* bv = (const float*)d_in[6];

    float* out     = (float*)d_out;                        // [B,S,E]
    float* weights = out + (long)BATCH * SEQ * EMB;        // [B,S,S]

    const long bse = (long)BATCH * SEQ * EMB;              // per-matrix elems
    float* Q = (float*)d_ws;                               // [B*S, E]
    float* Kt = Q + bse;                                   // [B*S, E]
    float* V = Kt + bse;                                   // [B*S, E]

    const float scale = 1.0f / sqrtf((float)EMB);
    const dim3 blk(256);

    // --- QKV projections: [B*S, E] x [E, E] + bias (ldb = EMB) ---
    {
        dim3 grid((BATCH * SEQ) / 128, EMB / 128, 1);
        wmma_gemm_128x128<false, true, false><<<grid, blk, 0, stream>>>(
            X, Wq, bq, Q, EMB, EMB, EMB, 0, 0, 0, 1.0f);
        wmma_gemm_128x128<false, true, false><<<grid, blk, 0, stream>>>(
            X, Wk, bk, Kt, EMB, EMB, EMB, 0, 0, 0, 1.0f);
        wmma_gemm_128x128<false, true, false><<<grid, blk, 0, stream>>>(
            X, Wv, bv, V, EMB, EMB, EMB, 0, 0, 0, 1.0f);
    }

    // --- scores = Q @ K^T * scale, per batch (B = K, ldb = EMB) ---
    {
        dim3 grid(SEQ / 128, SEQ / 128, BATCH);
        wmma_gemm_128x128<true, false, true><<<grid, blk, 0, stream>>>(
            Q, Kt, nullptr, weights,
            EMB, EMB, SEQ,
            (long)SEQ * EMB, (long)SEQ * EMB, (long)SEQ * SEQ, scale);
    }

    // --- softmax over last axis, in place in d_out weights region ---
    softmax_rows<<<BATCH * SEQ, 256, 0, stream>>>(weights);

    // --- output = weights @ V, per batch (B = V, ldb = EMB) ---
    {
        dim3 grid(SEQ / 128, EMB / 128, BATCH);
        wmma_gemm_128x128<false, false, false><<<grid, blk, 0, stream>>>(
            weights, V, nullptr, out,
            SEQ, SEQ, EMB,
            (long)SEQ * SEQ, (long)SEQ * EMB, (long)SEQ * EMB, 1.0f);
    }
}